// EntropyPatcher_4329327035038
// MI455X (gfx1250) — compile-verified
//
#include <hip/hip_runtime.h>
#include <hip/hip_bf16.h>

#define B_   256
#define L_   8192
#define E_   64
#define K_   5
#define PAD_ 4          // WINDOW // 2
#define THR_ 1.5f
#define PSH_ 3
#define PSL_ 12

typedef __attribute__((ext_vector_type(2))) float v2f;
typedef __attribute__((ext_vector_type(8))) float v8f;

// ---------------------------------------------------------------------------
// Kernel 1: sliding-window entropy, one thread per (row, col).
// Zero-padding in the reference's one-hot space == only count in-range
// positions; 'total' is the in-range window size (5..9).
// ---------------------------------------------------------------------------
__global__ void ep_entropy_kernel(const int* __restrict__ x,
                                  float* __restrict__ ent) {
  int idx = blockIdx.x * blockDim.x + threadIdx.x;
  if (idx >= B_ * L_) return;
  int row = idx >> 13;            // / L_
  int col = idx & (L_ - 1);       // % L_
  const int* xr = x + row * L_;

  int lo = col - PAD_; if (lo < 0) lo = 0;
  int hi = col + PAD_; if (hi > L_ - 1) hi = L_ - 1;

  int cnt[K_] = {0, 0, 0, 0, 0};
  int total = 0;
  for (int t = lo; t <= hi; ++t) {
    int v = xr[t];
    v = v < 0 ? 0 : (v > K_ - 1 ? K_ - 1 : v);
    ++cnt[v];
    ++total;
  }
  float inv_tot = 1.0f / (float)total;
  float e = 0.0f;
#pragma unroll
  for (int k = 0; k < K_; ++k) {
    float p = (float)cnt[k] * inv_tot;
    e -= p * __log2f(p + 1e-12f);   // v_log_f32 transcendental
  }
  ent[idx] = e;
}

// ---------------------------------------------------------------------------
// Kernel 2: serial segmentation chain, one WAVE per row.
// Lane l accumulates hidden dims l and l+32 of sum_p relu(mean_p*w1 + b1).
// Patch sums are exact integer sums => identical to the reference's
// prefix-sum difference. h_avg = h_sum / n_valid goes to workspace.
// ---------------------------------------------------------------------------
__global__ void ep_scan_kernel(const int* __restrict__ x,
                               const float* __restrict__ ent,
                               const float* __restrict__ w1,
                               const float* __restrict__ b1,
                               float* __restrict__ h_avg) {
  int gtid = blockIdx.x * blockDim.x + threadIdx.x;
  int wave = gtid >> 5;
  int lane = gtid & 31;
  if (wave >= B_) return;

  const int*   xr = x   + wave * L_;
  const float* er = ent + wave * L_;

  float w1l = w1[lane],       b1l = b1[lane];
  float w1h = w1[lane + 32],  b1h = b1[lane + 32];

  float h0 = 0.0f, h1 = 0.0f;
  int i = 0, nv = 0;
  while (i < L_) {
    float e  = er[i];                       // uniform across the wave
    int   ps = (e > THR_) ? PSH_ : PSL_;
    int   j  = i + ps; if (j > L_) j = L_;
    int s = 0;
    for (int t = i; t < j; ++t) s += xr[t]; // <=12 broadcast loads (L2-hot)
    float mean = (float)s / (float)(j - i);
    h0 += fmaxf(fmaf(mean, w1l, b1l), 0.0f);
    h1 += fmaxf(fmaf(mean, w1h, b1h), 0.0f);
    ++nv;
    i = j;
  }
  float inv = 1.0f / (float)nv;
  h_avg[wave * E_ + lane]      = h0 * inv;
  h_avg[wave * E_ + lane + 32] = h1 * inv;
}

// ---------------------------------------------------------------------------
// Kernel 3: blt = h_avg(256x64) @ w2^T(64x64) + b2, via f32 WMMA 16x16x4.
// One wave per 16x16 output tile; 64 tiles; 16 WMMA ops/tile over K=64.
// A layout (16x4 f32): lanes 0-15 hold M=lane, VGPRs = K{0,1}; lanes 16-31
// hold M=lane-16, VGPRs = K{2,3}. B (4x16) mirrored: N on lane[3:0], K pair
// split on lane halves. C/D: VGPR v -> row v (+8 for upper lane half).
// ---------------------------------------------------------------------------
__global__ void ep_gemm_kernel(const float* __restrict__ h_avg,
                               const float* __restrict__ w2,
                               const float* __restrict__ b2,
                               float* __restrict__ out) {
  int gtid = blockIdx.x * blockDim.x + threadIdx.x;
  int wave = gtid >> 5;          // 0..63
  int lane = gtid & 31;
  int tm = wave >> 2;            // 0..15 -> rows
  int tn = wave & 3;             // 0..3  -> cols
  int rb = tm * 16;
  int cb = tn * 16;
  int half = lane >> 4;          // 0 or 1
  int l15  = lane & 15;

  v8f c = {};
#pragma unroll
  for (int k0 = 0; k0 < E_; k0 += 4) {
    int ka = k0 + half * 2;
    v2f a, b;
    a.x = h_avg[(rb + l15) * E_ + ka];
    a.y = h_avg[(rb + l15) * E_ + ka + 1];
    b.x = w2[(cb + l15) * E_ + ka];        // B[k][n] = w2[n][k] (transpose)
    b.y = w2[(cb + l15) * E_ + ka + 1];
    c = __builtin_amdgcn_wmma_f32_16x16x4_f32(
        /*neg_a=*/false, a, /*neg_b=*/false, b,
        /*c_mod=*/(short)0, c, /*reuse_a=*/false, /*reuse_b=*/false);
  }

  float bias = b2[cb + l15];
#pragma unroll
  for (int v = 0; v < 8; ++v) {
    int m = v + half * 8;
    out[(rb + m) * E_ + cb + l15] = c[v] + bias;
  }
}

// ---------------------------------------------------------------------------
extern "C" void kernel_launch(void* const* d_in, const int* in_sizes, int n_in,
                              void* d_out, int out_size, void* d_ws, size_t ws_size,
                              hipStream_t stream) {
  (void)in_sizes; (void)n_in; (void)out_size; (void)ws_size;

  const int*   x  = (const int*)  d_in[0];   // [B, L] int32
  const float* w1 = (const float*)d_in[1];   // [E, 1]
  const float* b1 = (const float*)d_in[2];   // [E]
  const float* w2 = (const float*)d_in[3];   // [E, E]
  const float* b2 = (const float*)d_in[4];   // [E]

  float* blt = (float*)d_out;                // [B, E]   (output 0)
  float* ent = blt + B_ * E_;                // [B, L]   (output 1)
  float* h_avg = (float*)d_ws;               // [B, E] scratch = 64 KB

  // Phase 1: entropy (writes output 1 directly)
  ep_entropy_kernel<<<(B_ * L_ + 255) / 256, 256, 0, stream>>>(x, ent);

  // Phase 2: per-row serial scan, 1 wave/row -> 256 waves
  ep_scan_kernel<<<(B_ * 32 + 255) / 256, 256, 0, stream>>>(x, ent, w1, b1, h_avg);

  // Phase 3: 256x64x64 f32 WMMA GEMM + bias (writes output 0)
  ep_gemm_kernel<<<8, 256, 0, stream>>>(h_avg, w2, b2, blt);
}